// CliffordEngine_13383118094477
// MI455X (gfx1250) — compile-verified
//
#include <hip/hip_runtime.h>

typedef float v8f __attribute__((ext_vector_type(8)));
typedef float v2f __attribute__((ext_vector_type(2)));
typedef unsigned int uint_t;

#define WAVES_PER_BLOCK 8

// Octonion multiply e_i * e_j (i in 1..7, j in 0..7) -> (k, sign): e_i e_j = s * e_k
__device__ __forceinline__ void oct_mul(int i, int j, int& k, int& s) {
  if (j == 0) { k = i; s = 1; return; }
  if (i == j) { k = 0; s = -1; return; }
  // standard oriented Fano triples (index-doubling convention): e_a e_b = e_c cyclically
  const int T[7][3] = {{1,2,4},{2,3,5},{3,4,6},{4,5,7},{5,6,1},{6,7,2},{7,1,3}};
  for (int t = 0; t < 7; ++t) {
    int a = T[t][0], b = T[t][1], c = T[t][2];
    if (i == a && j == b) { k = c; s = 1;  return; }
    if (i == b && j == c) { k = a; s = 1;  return; }
    if (i == c && j == a) { k = b; s = 1;  return; }
    if (i == b && j == a) { k = c; s = -1; return; }
    if (i == c && j == b) { k = a; s = -1; return; }
    if (i == a && j == c) { k = b; s = -1; return; }
  }
  k = 0; s = 1; // unreachable
}

// One wave32 per multivector m. x' = phi^-1( phi(R) * phi(x) * phi(R~) ),
// phi : Cl(8,0) -> M(16,R), each 16x16x16 product = 4x v_wmma_f32_16x16x4_f32.
__global__ __launch_bounds__(256) void rotor_sandwich_cl8(
    const float* __restrict__ R, const float* __restrict__ X,
    const float* __restrict__ rev, float* __restrict__ out, int total_mv) {
  // ---- tables (built per block, LDS-resident) ----
  __shared__ unsigned char eRow[8][16];  __shared__ signed char eSgn[8][16];
  // encode table: (r,c) -> 16 entries u32 = (sign<<31) | (blade*4); b128-loadable
  __shared__ uint4 inv4[16][16][4];
  // decode table: blade a -> 16 entries u16 = ((row*16+c)*4) | (sign<<15)
  __shared__ unsigned short dec[256][16];
  // ---- per-wave staging ----
  __shared__ float Rs[WAVES_PER_BLOCK][256];
  __shared__ float Xs[WAVES_PER_BLOCK][256];
  __shared__ float Qs[WAVES_PER_BLOCK][256];  // R~ = R * reverse_signs
  __shared__ float Ms[WAVES_PER_BLOCK][256];  // 16x16 matrix scratch

  const int tid = threadIdx.x;

  // Phase 0a: Clifford generators of Cl(8,0) on R^16 as signed permutations.
  // E1 = sx (x) I8 ; E2 = sz (x) I8 ; E(2+i) = (sx*sz) (x) L_{e_i}, i=1..6.
  if (tid < 128) {
    int k = tid >> 4, c = tid & 15, r, s;
    if (k == 0)      { r = c ^ 8; s = 1; }
    else if (k == 1) { r = c; s = (c < 8) ? 1 : -1; }
    else {
      int i = k - 1, c0 = c & 7, cb = c >> 3, orow, osgn;
      oct_mul(i, c0, orow, osgn);
      if (cb == 0) { r = orow + 8; s = osgn; }     // (sx sz) block [1][0] = +L
      else         { r = orow;     s = -osgn; }    // (sx sz) block [0][1] = -L
    }
    eRow[k][c] = (unsigned char)r; eSgn[k][c] = (signed char)s;
  }
  __syncthreads();

  // Phase 0b: blade matrices Gamma_a = E_{i1} E_{i2} ... (ascending bits);
  // apply rightmost (largest bit) first; emit packed decode entries directly.
  {
    const int a = tid;
    for (int c = 0; c < 16; ++c) {
      int r = c, s = 1;
      for (int i = 7; i >= 0; --i)
        if ((a >> i) & 1) { int sn = eSgn[i][r]; r = eRow[i][r]; s *= sn; }
      dec[a][c] = (unsigned short)(((r * 16 + c) << 2) | ((s < 0) ? 0x8000 : 0));
    }
  }
  __syncthreads();

  // Phase 0c: invert: for each matrix entry (r,c), the 16 blades feeding it.
  {
    const int r = tid >> 4, c = tid & 15;
    uint_t* invU = (uint_t*)inv4;
    int cnt = 0;
    for (int a = 0; a < 256 && cnt < 16; ++a) {
      unsigned int e = dec[a][c];
      if (((e >> 6) & 0xF) == (unsigned)r) {        // row(a,c) == r ?
        invU[(r * 16 + c) * 16 + cnt] =
            ((uint_t)a << 2) | ((e & 0x8000u) << 16);
        ++cnt;
      }
    }
  }
  __syncthreads();

  const int wave = tid >> 5, lane = tid & 31;
  int m = blockIdx.x * WAVES_PER_BLOCK + wave;
  const bool live = (m < total_mv);
  if (!live) m = 0;                       // clamp loads; store is guarded
  const int q = lane & 15, h = lane >> 4; // WMMA lane coords

  // Phase 1: stage this wave's multivectors into LDS (coalesced).
  for (int v = 0; v < 8; ++v) {
    int idx = v * 32 + lane;
    float r = R[(size_t)m * 256 + idx];
    float x = X[(size_t)m * 256 + idx];
    Rs[wave][idx] = r;
    Xs[wave][idx] = x;
    Qs[wave][idx] = r * rev[idx];
  }
  __syncthreads();

  // encode: phi(vec)[r][c] = 16-term gather; sign applied by XOR on bit 31
  auto enc = [&](const uint_t* vecU, int r, int c) -> float {
    const uint4* tp = &inv4[r][c][0];
    float acc = 0.f;
#pragma unroll
    for (int g = 0; g < 4; ++g) {
      uint4 e = tp[g];                               // ds_load_b128
      uint_t v0 = vecU[(e.x & 0x3FCu) >> 2] ^ (e.x & 0x80000000u);
      uint_t v1 = vecU[(e.y & 0x3FCu) >> 2] ^ (e.y & 0x80000000u);
      uint_t v2 = vecU[(e.z & 0x3FCu) >> 2] ^ (e.z & 0x80000000u);
      uint_t v3 = vecU[(e.w & 0x3FCu) >> 2] ^ (e.w & 0x80000000u);
      acc += (__uint_as_float(v0) + __uint_as_float(v1)) +
             (__uint_as_float(v2) + __uint_as_float(v3));
    }
    return acc;
  };

  const uint_t* RsU = (const uint_t*)&Rs[wave][0];
  const uint_t* XsU = (const uint_t*)&Xs[wave][0];
  const uint_t* QsU = (const uint_t*)&Qs[wave][0];

  // Phase 2: P = phi(R) * phi(x)  (4x wmma f32 16x16x4, K tiled)
  v8f accP = {};
#pragma unroll
  for (int kk = 0; kk < 4; ++kk) {
    v2f a, b;
#pragma unroll
    for (int v = 0; v < 2; ++v) {
      int kl = kk * 4 + v + 2 * h;            // A: lane=M+16*(K>=2), vgpr=K&1
      a[v] = enc(RsU, q, kl);                 // phi(R)[M=q][K=kl]
      b[v] = enc(XsU, kl, q);                 // phi(x)[K=kl][N=q]
    }
    accP = __builtin_amdgcn_wmma_f32_16x16x4_f32(false, a, false, b,
                                                 (short)0, accP, false, false);
  }

  // Phase 3: re-layout P (C-layout -> A-layout) via LDS, then out = P * phi(R~)
#pragma unroll
  for (int v = 0; v < 8; ++v) Ms[wave][(v + 8 * h) * 16 + q] = accP[v];
  __syncthreads();

  v8f accO = {};
#pragma unroll
  for (int kk = 0; kk < 4; ++kk) {
    v2f a, b;
#pragma unroll
    for (int v = 0; v < 2; ++v) {
      int kl = kk * 4 + v + 2 * h;
      a[v] = Ms[wave][q * 16 + kl];           // P[M=q][K=kl]
      b[v] = enc(QsU, kl, q);                 // phi(R~)[K=kl][N=q]
    }
    accO = __builtin_amdgcn_wmma_f32_16x16x4_f32(false, a, false, b,
                                                 (short)0, accO, false, false);
  }
  __syncthreads();
#pragma unroll
  for (int v = 0; v < 8; ++v) Ms[wave][(v + 8 * h) * 16 + q] = accO[v];
  __syncthreads();

  // Phase 4: decode x'[a] = (1/16) tr(Gamma_a^T M): 16-term gather, XOR signs
  const uint_t* MsU = (const uint_t*)&Ms[wave][0];
  for (int v = 0; v < 8; ++v) {
    int a = v * 32 + lane;
    const uint_t* dp = (const uint_t*)&dec[a][0];   // 8 packed u16 pairs
    float acc = 0.f;
#pragma unroll
    for (int g = 0; g < 8; ++g) {
      uint_t e = dp[g];
      uint_t lo = e & 0xFFFFu;
      uint_t v0 = MsU[(lo & 0xFFCu) >> 2] ^ ((lo & 0x8000u) << 16);
      uint_t v1 = MsU[((e >> 16) & 0xFFCu) >> 2] ^ (e & 0x80000000u);
      acc += __uint_as_float(v0) + __uint_as_float(v1);
    }
    if (live) out[(size_t)m * 256 + a] = acc * 0.0625f;
  }
}

extern "C" void kernel_launch(void* const* d_in, const int* in_sizes, int n_in,
                              void* d_out, int out_size, void* d_ws, size_t ws_size,
                              hipStream_t stream) {
  (void)n_in; (void)d_ws; (void)ws_size; (void)out_size;
  const float* R   = (const float*)d_in[0];  // (B,S,256)
  const float* X   = (const float*)d_in[1];  // (B,S,256)
  // d_in[2] = dense cayley tensor: deliberately unused (64 MB of mostly zeros)
  const float* rev = (const float*)d_in[3];  // (256,)
  float* out = (float*)d_out;

  const int total_mv = in_sizes[0] / 256;                    // 2048
  const int blocks = (total_mv + WAVES_PER_BLOCK - 1) / WAVES_PER_BLOCK;
  rotor_sandwich_cl8<<<blocks, 32 * WAVES_PER_BLOCK, 0, stream>>>(
      R, X, rev, out, total_mv);
}